// GTN_2843268349956
// MI455X (gfx1250) — compile-verified
//
#include <hip/hip_runtime.h>
#include <math.h>

typedef __attribute__((ext_vector_type(2))) float v2f;
typedef __attribute__((ext_vector_type(8))) float v8f;

constexpr int  N_   = 2048;
constexpr int  C_   = 2;
constexpr int  WOUT_= 64;
constexpr int  B_   = 16;
constexpr int  S_   = 20;
constexpr int  U_   = 256;
constexpr int  I_   = 2048;
constexpr long long NN_ = (long long)N_ * N_;

// ---------------------------------------------------------------- async LDS
#if defined(__has_builtin)
#if __has_builtin(__builtin_amdgcn_global_load_async_to_lds_b128)
#define ASYNC_LDS 1
#endif
#endif
#ifndef ASYNC_LDS
#define ASYNC_LDS 0
#endif

typedef int v4i __attribute__((vector_size(16)));
typedef __attribute__((address_space(1))) v4i gv4i_t;   // global int4
typedef __attribute__((address_space(3))) v4i sv4i_t;   // LDS int4

__device__ __forceinline__ void async_copy16(const float* g, float* l) {
#if ASYNC_LDS
  // GLOBAL_LOAD_ASYNC_TO_LDS_B128: copies 16 bytes from global address g to LDS address l, tracked by ASYNCcnt
  __builtin_amdgcn_global_load_async_to_lds_b128((gv4i_t*)g, (sv4i_t*)l, 0, 0);
#else
  *(float4*)l = *(const float4*)g;  // synchronous fallback
#endif
}
__device__ __forceinline__ void wait_async_le8() {
#if ASYNC_LDS
  asm volatile("s_wait_asynccnt 0x8" ::: "memory");
#endif
}
__device__ __forceinline__ void wait_async_le0() {
#if ASYNC_LDS
  asm volatile("s_wait_asynccnt 0x0" ::: "memory");
#endif
}

// ---------------------------------------------------------------- WMMA helper
__device__ __forceinline__ v8f wmma_f32_4(v2f a, v2f b, v8f c) {
  // V_WMMA_F32_16X16X4_F32: (neg_a, A, neg_b, B, c_mod, C, reuse_a, reuse_b)
  return __builtin_amdgcn_wmma_f32_16x16x4_f32(false, a, false, b, (short)0, c,
                                               false, false);
}

// ---------------------------------------------------------------- softmax(w)
__global__ void softmax3_kernel(const float* __restrict__ w1a,
                                const float* __restrict__ w1b,
                                const float* __restrict__ w2,
                                float* __restrict__ sw) {
  int t = threadIdx.x;
  if (t >= 3 * C_) return;
  const float* srcs[3] = {w1a, w1b, w2};
  int m = t >> 1, c = t & 1;
  const float* w = srcs[m] + c * 4;
  float mx = fmaxf(fmaxf(w[0], w[1]), fmaxf(w[2], w[3]));
  float e0 = expf(w[0] - mx), e1 = expf(w[1] - mx);
  float e2 = expf(w[2] - mx), e3 = expf(w[3] - mx);
  float inv = 1.0f / (e0 + e1 + e2 + e3);
  float* dst = sw + m * 8 + c * 4;
  dst[0] = e0 * inv; dst[1] = e1 * inv; dst[2] = e2 * inv; dst[3] = e3 * inv;
}

// ---------------------------------------------------------------- gtconv
__global__ void gtconv_kernel(const float* __restrict__ A,
                              const float* __restrict__ sw,
                              float* __restrict__ Q) {
  long long i = (long long)blockIdx.x * blockDim.x + threadIdx.x;
  if (i >= NN_) return;
  float4 a = *(const float4*)(A + i * 4);
#pragma unroll
  for (int c = 0; c < C_; ++c) {
    float v = sw[c * 4 + 0] * a.x + sw[c * 4 + 1] * a.y +
              sw[c * 4 + 2] * a.z + sw[c * 4 + 3] * a.w;
    Q[(long long)c * NN_ + i] = v;
  }
}

// ---------------------------------------------------------------- big GEMM
// C[c] = A[c] @ B[c], 2048x2048x2048 per channel, fp32 WMMA.
// 128x128 tile/block, 8 waves (2x4), 64x32 per wave.
// Double-buffered LDS, async global->LDS staging (ASYNCcnt pipelined).
__global__ void __launch_bounds__(256)
wmma_gemm_nt_async(const float* __restrict__ Aall,
                   const float* __restrict__ Ball,
                   float* __restrict__ Call,
                   long long strideA, long long strideB, long long strideC) {
  constexpr int TM = 128, TN = 128, TK = 32;
  constexpr int NT = N_ / TK;                 // 64 k-tiles
  constexpr int ASTR = 36, BSTR = 136;        // padded, 16B-aligned strides
  extern __shared__ float smem[];
  float* As = smem;                           // [2][TM][ASTR]
  float* Bs = smem + 2 * TM * ASTR;           // [2][TK][BSTR]

  const float* A = Aall + (long long)blockIdx.z * strideA;
  const float* B = Ball + (long long)blockIdx.z * strideB;
  float*       C = Call + (long long)blockIdx.z * strideC;

  const int tid   = threadIdx.x;
  const int lane  = tid & 31;
  const int wave  = tid >> 5;
  const int waveM = wave >> 2;   // 0..1
  const int waveN = wave & 3;    // 0..3
  const int half  = lane >> 4;   // K-half select per A/B fragment layout
  const int l16   = lane & 15;
  const int rowBase = blockIdx.y * TM;
  const int colBase = blockIdx.x * TN;

  v8f acc[4][2];
#pragma unroll
  for (int mi = 0; mi < 4; ++mi)
#pragma unroll
    for (int ni = 0; ni < 2; ++ni) acc[mi][ni] = {};

  auto stage = [&](int k0, int buf) {
#pragma unroll
    for (int it = 0; it < 4; ++it) {          // A tile: 1024 float4
      int e4 = it * 256 + tid;
      int ra = e4 >> 3, ka = (e4 & 7) * 4;
      async_copy16(A + (long long)(rowBase + ra) * N_ + k0 + ka,
                   &As[(buf * TM + ra) * ASTR + ka]);
    }
#pragma unroll
    for (int it = 0; it < 4; ++it) {          // B tile: 1024 float4
      int e4 = it * 256 + tid;
      int kb = e4 >> 5, cb = (e4 & 31) * 4;
      async_copy16(B + (long long)(k0 + kb) * N_ + colBase + cb,
                   &Bs[(buf * TK + kb) * BSTR + cb]);
    }
  };

  stage(0, 0);
  stage(TK, 1);

  for (int t = 0; t < NT; ++t) {
    const int cur = t & 1;
    if (t < NT - 1) wait_async_le8(); else wait_async_le0();
    __syncthreads();

    const float* Asb = &As[cur * TM * ASTR];
    const float* Bsb = &Bs[cur * TK * BSTR];
#pragma unroll
    for (int kk = 0; kk < TK; kk += 4) {
      v2f afr[4], bfr[2];
#pragma unroll
      for (int mi = 0; mi < 4; ++mi) {
        int r = waveM * 64 + mi * 16 + l16;
        v2f a;
        a.x = Asb[r * ASTR + kk + half * 2];
        a.y = Asb[r * ASTR + kk + half * 2 + 1];
        afr[mi] = a;
      }
#pragma unroll
      for (int ni = 0; ni < 2; ++ni) {
        int cc = waveN * 32 + ni * 16 + l16;
        v2f b;
        b.x = Bsb[(kk + half * 2) * BSTR + cc];
        b.y = Bsb[(kk + half * 2 + 1) * BSTR + cc];
        bfr[ni] = b;
      }
#pragma unroll
      for (int mi = 0; mi < 4; ++mi)
#pragma unroll
        for (int ni = 0; ni < 2; ++ni)
          acc[mi][ni] = wmma_f32_4(afr[mi], bfr[ni], acc[mi][ni]);
    }
    __syncthreads();
    if (t + 2 < NT) stage((t + 2) * TK, cur);   // refill freed buffer
  }

  // epilogue: VGPR r -> M = r (lanes 0-15) / 8+r (lanes 16-31)
#pragma unroll
  for (int mi = 0; mi < 4; ++mi) {
#pragma unroll
    for (int ni = 0; ni < 2; ++ni) {
      int colg = colBase + waveN * 32 + ni * 16 + l16;
#pragma unroll
      for (int r = 0; r < 8; ++r) {
        int rowg = rowBase + waveM * 64 + mi * 16 + half * 8 + r;
        C[(long long)rowg * N_ + colg] = acc[mi][ni][r];
      }
    }
  }
}

// ---------------------------------------------------------------- small GEMM
// Xc[c] = relu(Hg[c]^T @ Xw): TRANSA, N-dim 64, synchronous LDS staging.
template <bool TRANSA, bool RELU, int NCOLS, int LDA, int LDB, int LDC>
__global__ void wmma_gemm_kernel(const float* __restrict__ Aall,
                                 const float* __restrict__ Ball,
                                 float* __restrict__ Call,
                                 long long strideA, long long strideB,
                                 long long strideC) {
  constexpr int TM = 128, TN = 128, TK = 32, KDIM = N_;
  __shared__ float As[TM][36];
  __shared__ float Bs[TK][136];

  const float* A = Aall + (long long)blockIdx.z * strideA;
  const float* B = Ball + (long long)blockIdx.z * strideB;
  float*       C = Call + (long long)blockIdx.z * strideC;

  const int tid   = threadIdx.x;
  const int lane  = tid & 31;
  const int wave  = tid >> 5;
  const int waveM = wave >> 2;
  const int waveN = wave & 3;
  const int half  = lane >> 4;
  const int l16   = lane & 15;
  const int rowBase = blockIdx.y * TM;
  const int colBase = blockIdx.x * TN;

  v8f acc[4][2];
#pragma unroll
  for (int mi = 0; mi < 4; ++mi)
#pragma unroll
    for (int ni = 0; ni < 2; ++ni) acc[mi][ni] = {};

  for (int k0 = 0; k0 < KDIM; k0 += TK) {
    if (!TRANSA) {
#pragma unroll
      for (int it = 0; it < 4; ++it) {
        int e4 = it * 256 + tid;
        int r  = e4 >> 3;
        int kk = (e4 & 7) * 4;
        float4 v = *(const float4*)(A + (long long)(rowBase + r) * LDA + k0 + kk);
        As[r][kk + 0] = v.x; As[r][kk + 1] = v.y;
        As[r][kk + 2] = v.z; As[r][kk + 3] = v.w;
      }
    } else {
#pragma unroll
      for (int it = 0; it < 4; ++it) {
        int e4 = it * 256 + tid;
        int kk = e4 >> 5;
        int r  = (e4 & 31) * 4;
        float4 v = *(const float4*)(A + (long long)(k0 + kk) * LDA + rowBase + r);
        As[r + 0][kk] = v.x; As[r + 1][kk] = v.y;
        As[r + 2][kk] = v.z; As[r + 3][kk] = v.w;
      }
    }
#pragma unroll
    for (int it = 0; it < 4; ++it) {
      int e4 = it * 256 + tid;
      int kk = e4 >> 5;
      int cg = (e4 & 31) * 4;
      float4 v;
      if (colBase + cg < NCOLS) {
        v = *(const float4*)(B + (long long)(k0 + kk) * LDB + colBase + cg);
      } else {
        v.x = v.y = v.z = v.w = 0.0f;
      }
      Bs[kk][cg + 0] = v.x; Bs[kk][cg + 1] = v.y;
      Bs[kk][cg + 2] = v.z; Bs[kk][cg + 3] = v.w;
    }
    __syncthreads();

#pragma unroll
    for (int kk = 0; kk < TK; kk += 4) {
      v2f afr[4], bfr[2];
#pragma unroll
      for (int mi = 0; mi < 4; ++mi) {
        int r = waveM * 64 + mi * 16 + l16;
        v2f a; a.x = As[r][kk + half * 2]; a.y = As[r][kk + half * 2 + 1];
        afr[mi] = a;
      }
#pragma unroll
      for (int ni = 0; ni < 2; ++ni) {
        int cc = waveN * 32 + ni * 16 + l16;
        v2f b; b.x = Bs[kk + half * 2][cc]; b.y = Bs[kk + half * 2 + 1][cc];
        bfr[ni] = b;
      }
#pragma unroll
      for (int mi = 0; mi < 4; ++mi)
#pragma unroll
        for (int ni = 0; ni < 2; ++ni)
          acc[mi][ni] = wmma_f32_4(afr[mi], bfr[ni], acc[mi][ni]);
    }
    __syncthreads();
  }

#pragma unroll
  for (int mi = 0; mi < 4; ++mi) {
#pragma unroll
    for (int ni = 0; ni < 2; ++ni) {
      int colg = colBase + waveN * 32 + ni * 16 + l16;
      if (colg < NCOLS) {
#pragma unroll
        for (int r = 0; r < 8; ++r) {
          int rowg = rowBase + waveM * 64 + mi * 16 + half * 8 + r;
          float v = acc[mi][ni][r];
          if (RELU) v = fmaxf(v, 0.0f);
          C[(long long)rowg * LDC + colg] = v;
        }
      }
    }
  }
}

// ---------------------------------------------------------------- norm: dinv
__global__ void colsum_kernel(const float* __restrict__ H,
                              float* __restrict__ dinv, int addDiag) {
  int m = blockIdx.x * blockDim.x + threadIdx.x;
  int c = blockIdx.y;
  const float* Hc = H + (long long)c * NN_;
  float s = 0.0f;
  for (int n = 0; n < N_; ++n) s += Hc[(long long)n * N_ + m];
  s -= Hc[(long long)m * N_ + m];
  if (addDiag) s += 1.0f;
  dinv[c * N_ + m] = (s == 0.0f) ? 0.0f : 1.0f / s;
}

__global__ void normscale_kernel(const float* __restrict__ H,
                                 const float* __restrict__ dinv,
                                 float* __restrict__ Out, int addDiag) {
  long long i = (long long)blockIdx.x * blockDim.x + threadIdx.x;
  int c = blockIdx.y;
  if (i >= NN_) return;
  int n = (int)(i >> 11);
  int m = (int)(i & (N_ - 1));
  float v = (n == m) ? (addDiag ? 1.0f : 0.0f) : H[(long long)c * NN_ + i];
  Out[(long long)c * NN_ + i] = v * dinv[c * N_ + m];
}

// ---------------------------------------------------------------- Xw = X@gcn_w
__global__ void xw_kernel(const float* __restrict__ X,
                          const float* __restrict__ Wg,
                          float* __restrict__ Xw) {
  int idx = blockIdx.x * blockDim.x + threadIdx.x;
  int n = idx >> 6, w = idx & 63;
  float s = 0.0f;
  for (int j = 0; j < 64; ++j) s += X[n * 64 + j] * Wg[j * 64 + w];
  Xw[idx] = s;
}

// ---------------------------------------------------------------- lin1
__global__ void lin1_kernel(const float* __restrict__ Xc,
                            const float* __restrict__ W,
                            const float* __restrict__ bias,
                            float* __restrict__ Xo) {
  int idx = blockIdx.x * blockDim.x + threadIdx.x;
  int n = idx >> 6, o = idx & 63;
  float s = bias[o];
#pragma unroll
  for (int c = 0; c < C_; ++c)
    for (int w = 0; w < 64; ++w)
      s += Xc[((long long)c * N_ + n) * 64 + w] * W[o * 128 + c * 64 + w];
  Xo[idx] = s;
}

// ---------------------------------------------------------------- basket pool
__global__ void pool_kernel(const int* __restrict__ seqs,
                            const float* __restrict__ Xo,
                            float* __restrict__ basket) {
  int bs = blockIdx.x;
  int w  = threadIdx.x;
  const int* row = seqs + (long long)bs * I_;
  float m = -__builtin_inff();
  int any = 0;
  for (int i = 0; i < I_; ++i) {
    if (row[i] != 0) { any = 1; m = fmaxf(m, Xo[i * 64 + w]); }
  }
  basket[bs * 64 + w] = any ? m : 0.0f;
}

// ---------------------------------------------------------------- LSTM
__global__ void init_hc_kernel(const float* __restrict__ h0,
                               const float* __restrict__ c0,
                               float* __restrict__ h, float* __restrict__ c) {
  int i = blockIdx.x * blockDim.x + threadIdx.x;
  if (i < B_ * U_) { h[i] = h0[i]; c[i] = c0[i]; }
}

__global__ void lstm_step_kernel(const float* __restrict__ basket,
                                 const float* __restrict__ wih,
                                 const float* __restrict__ whh,
                                 const float* __restrict__ bih,
                                 const float* __restrict__ bhh,
                                 float* __restrict__ h, float* __restrict__ c,
                                 float* __restrict__ lout, int s) {
  __shared__ float hs[U_];
  __shared__ float xs[WOUT_];
  int b = blockIdx.x, u = threadIdx.x;
  hs[u] = h[b * U_ + u];
  if (u < WOUT_) xs[u] = basket[(b * S_ + s) * WOUT_ + u];
  __syncthreads();
  float g[4];
#pragma unroll
  for (int gate = 0; gate < 4; ++gate) {
    int row = gate * U_ + u;
    float sum = bih[row] + bhh[row];
    const float* wi = wih + (long long)row * WOUT_;
    for (int j = 0; j < WOUT_; ++j) sum += wi[j] * xs[j];
    const float* wh = whh + (long long)row * U_;
    for (int j = 0; j < U_; ++j) sum += wh[j] * hs[j];
    g[gate] = sum;
  }
  float ig = 1.0f / (1.0f + expf(-g[0]));
  float fg = 1.0f / (1.0f + expf(-g[1]));
  float gg = tanhf(g[2]);
  float og = 1.0f / (1.0f + expf(-g[3]));
  float cn = fg * c[b * U_ + u] + ig * gg;
  float hn = og * tanhf(cn);
  c[b * U_ + u] = cn;
  h[b * U_ + u] = hn;
  lout[(b * S_ + s) * U_ + u] = hn;
}

// ---------------------------------------------------------------- output head
__global__ void out_kernel(const float* __restrict__ lout,
                           const int* __restrict__ seq_len,
                           const float* __restrict__ out_w,
                           float* __restrict__ out) {
  int idx = blockIdx.x * blockDim.x + threadIdx.x;
  if (idx >= B_ * I_) return;
  int b = idx / I_, i = idx % I_;
  int t = seq_len[b] - 1;
  t = t < 0 ? 0 : (t > S_ - 1 ? S_ - 1 : t);
  const float* a = lout + (b * S_ + t) * U_;
  const float* w = out_w + (long long)i * U_;
  float s = 0.0f;
  for (int j = 0; j < U_; ++j) s += a[j] * w[j];
  out[idx] = 1.0f / (1.0f + expf(-s));
}

// ---------------------------------------------------------------- launch
extern "C" void kernel_launch(void* const* d_in, const int* in_sizes, int n_in,
                              void* d_out, int out_size, void* d_ws,
                              size_t ws_size, hipStream_t stream) {
  (void)in_sizes; (void)n_in; (void)out_size; (void)ws_size;
  const float* A        = (const float*)d_in[0];
  const float* X        = (const float*)d_in[1];
  const int*   seq_len  = (const int*)d_in[2];
  const int*   seqs     = (const int*)d_in[3];
  const float* h0       = (const float*)d_in[4];
  const float* c0       = (const float*)d_in[5];
  const float* w1a      = (const float*)d_in[6];
  const float* w1b      = (const float*)d_in[7];
  const float* w2       = (const float*)d_in[8];
  const float* gcn_w    = (const float*)d_in[9];
  const float* lin1_w   = (const float*)d_in[10];
  const float* lin1_b   = (const float*)d_in[11];
  const float* lstm_wih = (const float*)d_in[12];
  const float* lstm_whh = (const float*)d_in[13];
  const float* lstm_bih = (const float*)d_in[14];
  const float* lstm_bhh = (const float*)d_in[15];
  const float* out_w    = (const float*)d_in[16];
  float* out = (float*)d_out;

  float* wsf  = (float*)d_ws;
  float* sw   = wsf;
  float* dinv = sw   + 64;
  float* Xw   = dinv + C_ * N_;
  float* Xc   = Xw   + N_ * 64;
  float* Xo   = Xc   + (long long)C_ * N_ * 64;
  float* bas  = Xo   + N_ * 64;
  float* hbuf = bas  + B_ * S_ * 64;
  float* cbuf = hbuf + B_ * U_;
  float* lout = cbuf + B_ * U_;
  float* buf0 = lout + B_ * S_ * U_;
  float* buf1 = buf0 + C_ * NN_;
  float* buf2 = buf1 + C_ * NN_;

  constexpr unsigned GEMM_LDS = (2 * 128 * 36 + 2 * 32 * 136) * sizeof(float);

  // 1) softmax of edge-mixing weights
  softmax3_kernel<<<1, 32, 0, stream>>>(w1a, w1b, w2, sw);

  // 2) Q1 = gtconv(w1a), Q2 = gtconv(w1b)
  int gtBlocks = (int)(NN_ / 256);
  gtconv_kernel<<<gtBlocks, 256, 0, stream>>>(A, sw + 0,  buf0);
  gtconv_kernel<<<gtBlocks, 256, 0, stream>>>(A, sw + 8,  buf1);

  // 3) H = Q1 @ Q2 (async double-buffered WMMA)
  dim3 gBig(N_ / 128, N_ / 128, C_);
  wmma_gemm_nt_async<<<gBig, 256, GEMM_LDS, stream>>>(buf0, buf1, buf2,
                                                      NN_, NN_, NN_);

  // 4) Hn = norm(H, add=false) -> buf0
  colsum_kernel<<<dim3(N_ / 256, C_), 256, 0, stream>>>(buf2, dinv, 0);
  normscale_kernel<<<dim3(gtBlocks, C_), 256, 0, stream>>>(buf2, dinv, buf0, 0);

  // 5) Q3 = gtconv(w2) -> buf1
  gtconv_kernel<<<gtBlocks, 256, 0, stream>>>(A, sw + 16, buf1);

  // 6) H2 = Hn @ Q3 -> buf2
  wmma_gemm_nt_async<<<gBig, 256, GEMM_LDS, stream>>>(buf0, buf1, buf2,
                                                      NN_, NN_, NN_);

  // 7) Hg = norm(H2, add=true) -> buf0
  colsum_kernel<<<dim3(N_ / 256, C_), 256, 0, stream>>>(buf2, dinv, 1);
  normscale_kernel<<<dim3(gtBlocks, C_), 256, 0, stream>>>(buf2, dinv, buf0, 1);

  // 8) Xw = X @ gcn_w
  xw_kernel<<<(N_ * 64) / 256, 256, 0, stream>>>(X, gcn_w, Xw);

  // 9) Xc = relu(Hg^T @ Xw)
  dim3 gSmall(1, N_ / 128, C_);
  wmma_gemm_kernel<true, true, WOUT_, N_, WOUT_, WOUT_>
      <<<gSmall, 256, 0, stream>>>(buf0, Xw, Xc, NN_, 0, (long long)N_ * 64);

  // 10) X_ = cat-channels @ lin1_w.T + b
  lin1_kernel<<<(N_ * 64) / 256, 256, 0, stream>>>(Xc, lin1_w, lin1_b, Xo);

  // 11) ragged basket max-pool
  pool_kernel<<<B_ * S_, 64, 0, stream>>>(seqs, Xo, bas);

  // 12) LSTM (20 sequential steps)
  init_hc_kernel<<<(B_ * U_) / 256, 256, 0, stream>>>(h0, c0, hbuf, cbuf);
  for (int s = 0; s < S_; ++s) {
    lstm_step_kernel<<<B_, U_, 0, stream>>>(bas, lstm_wih, lstm_whh, lstm_bih,
                                            lstm_bhh, hbuf, cbuf, lout, s);
  }

  // 13) gather last state, project, sigmoid
  out_kernel<<<(B_ * I_) / 256, 256, 0, stream>>>(lout, seq_len, out_w, out);
}